// BipartiteSAGEConv_61409442399048
// MI455X (gfx1250) — compile-verified
//
#include <hip/hip_runtime.h>
#include <math.h>

#define N_NODES 100000
#define N_EDGES 1600000
#define D       128
#define EPS     1e-12f

typedef __attribute__((ext_vector_type(2))) float v2f;
typedef __attribute__((ext_vector_type(8))) float v8f;

// ---------------------------------------------------------------------------
// Kernel 1: zero accumulator workspace (summed[N*D] followed by cnt[N])
// ---------------------------------------------------------------------------
__global__ void sage_zero_ws(float* __restrict__ ws, unsigned n4) {
    unsigned i = blockIdx.x * blockDim.x + threadIdx.x;
    if (i < n4) {
        float4 z = make_float4(0.f, 0.f, 0.f, 0.f);
        *(float4*)(ws + (size_t)i * 4) = z;
    }
}

// ---------------------------------------------------------------------------
// Kernel 2: edge scatter. One wave32 per edge: lanes cooperatively gather the
// 512B source row (float4 per lane) and atomically add into summed[dst].
// Working set (x: 51MB, summed: 51MB) is L2-resident on MI455X (192MB L2).
// ---------------------------------------------------------------------------
__global__ __launch_bounds__(256) void sage_scatter(
    const float* __restrict__ x,
    const int*   __restrict__ src,
    const int*   __restrict__ dst,
    float*       __restrict__ summed,
    float*       __restrict__ cnt) {
    unsigned tid  = blockIdx.x * blockDim.x + threadIdx.x;
    unsigned e    = tid >> 5;
    unsigned lane = tid & 31u;
    if (e >= N_EDGES) return;
    int s = src[e];
    int d = dst[e];
    float4 v = *(const float4*)(x + (size_t)s * D + lane * 4);
    float* p = summed + (size_t)d * D + lane * 4;
    atomicAdd(p + 0, v.x);
    atomicAdd(p + 1, v.y);
    atomicAdd(p + 2, v.z);
    atomicAdd(p + 3, v.w);
    if (lane == 0) atomicAdd(cnt + d, 1.0f);
}

// ---------------------------------------------------------------------------
// Kernel 3: fused  out = normalize( (summed/cnt)@W_l + b_l + x@W_r )
// One block = 16 node rows. 8 waves, wave w owns output columns [16w,16w+16).
// fp32 WMMA 16x16x4: A-frag (16x4) lanes 0-15 carry K=k,k+1 / lanes 16-31
// K=k+2,k+3 (v2f); B-frag (4x16) mirrored. C/D: v8f, row = vgpr + 8*(lane>>4),
// col = lane&15 within the tile.
// ---------------------------------------------------------------------------
__global__ __launch_bounds__(256) void sage_gemm_norm(
    const float* __restrict__ x,
    const float* __restrict__ summed,
    const float* __restrict__ cnt,
    const float* __restrict__ Wl,
    const float* __restrict__ bl,
    const float* __restrict__ Wr,
    float*       __restrict__ out) {
    __shared__ __align__(16) float tile[16 * 132];   // stride 132: 16B-aligned rows, conflict pad

    const int row0 = blockIdx.x * 16;
    const int wave = threadIdx.x >> 5;
    const int lane = threadIdx.x & 31;
    const int m    = lane & 15;      // A: matrix row   B/C/D: column within tile
    const int h    = lane >> 4;      // K-half select for A/B frags; +8 row offset for C/D
    const int n    = wave * 16 + m;  // global output column

    const float invc = 1.0f / fmaxf(cnt[row0 + m], 1.0f);
    const float* arow_s = summed + (size_t)(row0 + m) * D;
    const float* arow_x = x      + (size_t)(row0 + m) * D;

    v8f acc = {0.f, 0.f, 0.f, 0.f, 0.f, 0.f, 0.f, 0.f};

#pragma unroll 8
    for (int k = 0; k < D; k += 4) {
        const int kk = k + 2 * h;
        float2 ts = *(const float2*)(arow_s + kk);
        float2 tx = *(const float2*)(arow_x + kk);
        v2f am; am.x = ts.x * invc; am.y = ts.y * invc;   // mean path A-frag
        v2f ax; ax.x = tx.x;        ax.y = tx.y;          // root path A-frag
        v2f bL; bL.x = Wl[kk * D + n]; bL.y = Wl[(kk + 1) * D + n];
        v2f bR; bR.x = Wr[kk * D + n]; bR.y = Wr[(kk + 1) * D + n];
        acc = __builtin_amdgcn_wmma_f32_16x16x4_f32(false, am, false, bL,
                                                    (short)0, acc, false, false);
        acc = __builtin_amdgcn_wmma_f32_16x16x4_f32(false, ax, false, bR,
                                                    (short)0, acc, false, false);
    }

    // bias depends only on the column; stage biased tile to LDS
    const float bias = bl[n];
#pragma unroll
    for (int i = 0; i < 8; ++i) {
        tile[(i + 8 * h) * 132 + n] = acc[i] + bias;
    }
    __syncthreads();

    // Row-wise L2 normalize: thread t owns row r = t>>4, 8-col segment t&15.
    const int r   = threadIdx.x >> 4;
    const int seg = threadIdx.x & 15;
    const float* rp = tile + r * 132 + seg * 8;
    float4 v0 = *(const float4*)(rp + 0);
    float4 v1 = *(const float4*)(rp + 4);
    float ss = v0.x * v0.x + v0.y * v0.y + v0.z * v0.z + v0.w * v0.w +
               v1.x * v1.x + v1.y * v1.y + v1.z * v1.z + v1.w * v1.w;
    // reduce the 16 partials of each row (masks < 16 stay inside the 16-group)
#pragma unroll
    for (int mask = 1; mask < 16; mask <<= 1) ss += __shfl_xor(ss, mask, 32);
    const float inv = 1.0f / fmaxf(sqrtf(ss), EPS);

    float* op = out + (size_t)(row0 + r) * D + seg * 8;
    float4 o0 = make_float4(v0.x * inv, v0.y * inv, v0.z * inv, v0.w * inv);
    float4 o1 = make_float4(v1.x * inv, v1.y * inv, v1.z * inv, v1.w * inv);
    *(float4*)(op + 0) = o0;
    *(float4*)(op + 4) = o1;
}

// ---------------------------------------------------------------------------
extern "C" void kernel_launch(void* const* d_in, const int* in_sizes, int n_in,
                              void* d_out, int out_size, void* d_ws, size_t ws_size,
                              hipStream_t stream) {
    const float* x   = (const float*)d_in[0];
    const int*   ei  = (const int*)d_in[1];   // harness: integer -> const int*
    const float* Wl  = (const float*)d_in[2];
    const float* bl  = (const float*)d_in[3];
    const float* Wr  = (const float*)d_in[4];
    float*       out = (float*)d_out;

    float* summed = (float*)d_ws;                       // N*D floats
    float* cnt    = summed + (size_t)N_NODES * D;       // N floats

    const int* src = ei;            // edge_index[0]
    const int* dst = ei + N_EDGES;  // edge_index[1]

    // 1) zero summed + cnt  (N*(D+1) = 12,900,000 floats, divisible by 4)
    {
        unsigned n4 = (unsigned)((size_t)N_NODES * (D + 1) / 4);
        unsigned blocks = (n4 + 255u) / 256u;
        sage_zero_ws<<<blocks, 256, 0, stream>>>((float*)d_ws, n4);
    }
    // 2) scatter-add: one wave32 per edge
    {
        size_t threads = (size_t)N_EDGES * 32;
        unsigned blocks = (unsigned)((threads + 255) / 256);
        sage_scatter<<<blocks, 256, 0, stream>>>(x, src, dst, summed, cnt);
    }
    // 3) fused mean-div + dual GEMM (fp32 WMMA) + bias + L2 normalize
    {
        unsigned blocks = N_NODES / 16;  // 100000 = 16 * 6250 exactly
        sage_gemm_norm<<<blocks, 256, 0, stream>>>(x, summed, cnt, Wl, bl, Wr, out);
    }
}